// Discriminator_56590489092417
// MI455X (gfx1250) — compile-verified
//
#include <hip/hip_runtime.h>

// ---------------------------------------------------------------------------
// Stacked LSTM (3 layers) + Dense(1) for MI455X / gfx1250 (wave32, WMMA, TDM).
//
// Shapes: x[B=128,T=256,D=128]; per layer W[Din,1024], R[256,1024], b[1024];
// Wd[256,1], bd[1]; out[B,T,1].
//
//  * xz = in_seq @ W + b hoisted out of the scan as one big f32-WMMA GEMM
//    per layer. 4-wave blocks share a 16xK A-tile staged into LDS by the
//    Tensor Data Mover (tensor_load_to_lds + s_wait_tensorcnt).
//  * Recurrent scan: one launch per step; block = 4 waves = gates i,f,g,o.
//    h_prev 16x256 tile staged once per block via TDM (was 4x redundant
//    global reads on the latency-critical chain). Gates fused via LDS.
//    h double-buffered across steps; c in place.
//  * Dense: wave-per-output shuffle reduction.
//
// Workspace (floats): xz 32768*1024, hseqA/B 32768*256 each, states 3*32768
// => ~202 MB total in d_ws.
// ---------------------------------------------------------------------------

typedef __attribute__((ext_vector_type(2))) float    v2f;
typedef __attribute__((ext_vector_type(8))) float    v8f;
typedef __attribute__((ext_vector_type(4))) uint32_t v4u;
typedef __attribute__((ext_vector_type(8))) uint32_t v8u;

#define Bq 128
#define Tq 256
#define Uq 256
#define Gq 1024   // 4*U

// ---- TDM: 2D tile load Global -> LDS (D# built per CDNA5 ISA) --
// rows x cols f32 tile, row stride ld_elems, fully in-bounds.
__device__ __forceinline__ void tdm_load_2d_f32(const float* gptr, uint32_t lds_off,
                                                uint32_t rows, uint32_t cols,
                                                uint32_t ld_elems) {
  const uint64_t ga = (uint64_t)(uintptr_t)gptr;
  v4u g0;
  g0[0] = 1u;                                            // count=1 (user mode)
  g0[1] = lds_off;                                       // lds_addr (bytes)
  g0[2] = (uint32_t)ga;                                  // global_addr[31:0]
  g0[3] = ((uint32_t)(ga >> 32) & 0x01FFFFFFu) | (2u << 30); // addr[56:32]|type=2
  v8u g1;
  g1[0] = 2u << 16;                                      // data_size=4B; mask=0
  g1[1] = (cols & 0xFFFFu) << 16;                        // tensor_dim0[15:0]
  g1[2] = ((cols >> 16) & 0xFFFFu) | ((rows & 0xFFFFu) << 16); // dim0_hi|dim1_lo
  g1[3] = ((rows >> 16) & 0xFFFFu) | ((cols & 0xFFFFu) << 16); // dim1_hi|tile_dim0
  g1[4] = rows & 0xFFFFu;                                // tile_dim1; tile_dim2=0
  g1[5] = ld_elems;                                      // tensor_dim0_stride lo
  g1[6] = 0u;                                            // stride hi | dim1_stride lo
  g1[7] = 0u;
  const v4u gz = {0u, 0u, 0u, 0u};                       // groups 2/3 (2D tensor)
  asm volatile("tensor_load_to_lds %0, %1, %2, %3"
               :: "s"(g0), "s"(g1), "s"(gz), "s"(gz)
               : "memory");
}

// ---- f32 WMMA fragment helpers (16x16x4, wave32) --------------------------
// A (16x4): lane m=lane&15 holds A[m][kb], A[m][kb+1] with kb=2*(lane>>4).
// B (4x16): lane n=lane&15 holds B[kb][n], B[kb+1][n] with kb=2*(lane>>4).
// C/D (16x16): acc[r] at lane -> (M = r + 8*(lane>>4), N = lane&15).

__device__ __forceinline__ v2f load_a_lds(const float* hA, int ldA, int k, int lane) {
  const int m  = lane & 15;
  const int kb = (lane >> 4) << 1;
  const float* p = hA + m * ldA + k + kb;      // LDS after inlining -> ds_load_b64
  v2f a; a.x = p[0]; a.y = p[1];
  return a;
}

__device__ __forceinline__ v2f load_b_frag(const float* __restrict__ Bm,
                                           int ldB, int k, int col0, int lane) {
  const int n  = lane & 15;
  const int kb = (lane >> 4) << 1;
  v2f b;
  b.x = Bm[(size_t)(k + kb)     * ldB + col0 + n];
  b.y = Bm[(size_t)(k + kb + 1) * ldB + col0 + n];
  return b;
}

__device__ __forceinline__ v8f wmma_f32(v2f a, v2f b, v8f c) {
  // (neg_a, A, neg_b, B, c_mod, C, reuse_a, reuse_b)
  return __builtin_amdgcn_wmma_f32_16x16x4_f32(false, a, false, b,
                                               (short)0, c, false, false);
}

// ---- Kernel 1: xz = in_seq @ W + bias  ([M=B*T, K=Din] x [K, G]) ----------
// Block: 128 threads (4 waves), each wave one 16x16 N-tile; shared A tile
// (16 x K) staged to LDS once per block by the TDM.
__global__ __launch_bounds__(128) void gemm_xz_kernel(
    const float* __restrict__ A,     // [M, K] row-major
    const float* __restrict__ W,     // [K, G] row-major
    const float* __restrict__ bias,  // [G]
    float* __restrict__ xz,          // [M, G]
    int K) {
  __shared__ float hA[16 * 256];               // 16 x K tile (K <= 256)

  const int wave = threadIdx.x >> 5;
  const int lane = threadIdx.x & 31;
  const int m0 = blockIdx.x * 16;
  const int n0 = blockIdx.y * 64 + wave * 16;
  const int n  = lane & 15;
  const int hi = lane >> 4;

  if (threadIdx.x < 32) {
    tdm_load_2d_f32(A + (size_t)m0 * K, (uint32_t)(uintptr_t)hA, 16, K, K);
    __builtin_amdgcn_s_wait_tensorcnt(0);
  }
  __syncthreads();

  v8f acc;
  const float bv = bias[n0 + n];
#pragma unroll
  for (int r = 0; r < 8; ++r) acc[r] = bv;

  for (int k = 0; k < K; k += 4) {
    v2f a = load_a_lds(hA, K, k, lane);
    v2f b = load_b_frag(W, Gq, k, n0, lane);
    acc = wmma_f32(a, b, acc);
  }

#pragma unroll
  for (int r = 0; r < 8; ++r)
    xz[(size_t)(m0 + r + 8 * hi) * Gq + n0 + n] = acc[r];
}

// ---- Kernel 2: one LSTM time step -----------------------------------------
// Grid: (B/16, U/16); block: 128 threads (4 waves = gates i,f,g,o).
__global__ __launch_bounds__(128) void lstm_step_kernel(
    const float* __restrict__ xz,      // [B*T, G] (row = b*T + t)
    const float* __restrict__ R,       // [U, G]
    const float* __restrict__ h_prev,  // [B, U]
    float* __restrict__ h_next,        // [B, U]
    float* __restrict__ c,             // [B, U] (in-place)
    float* __restrict__ h_seq,         // [B, T, U]
    int t) {
  __shared__ float hA[16 * Uq];                // h_prev 16x256 tile (16 KB)
  __shared__ float zsh[4][16][16];             // gate exchange (4 KB)

  const int gate = threadIdx.x >> 5;           // 0..3 -> i,f,g,o
  const int lane = threadIdx.x & 31;
  const int m0 = blockIdx.x * 16;              // batch tile
  const int u0 = blockIdx.y * 16;              // unit tile
  const int col0 = gate * Uq + u0;             // z-column base for this gate
  const int n  = lane & 15;
  const int hi = lane >> 4;

  // Stage h_prev tile once per block via TDM (shared by all 4 gate-waves).
  if (threadIdx.x < 32) {
    tdm_load_2d_f32(h_prev + (size_t)m0 * Uq, (uint32_t)(uintptr_t)hA, 16, Uq, Uq);
    __builtin_amdgcn_s_wait_tensorcnt(0);
  }

  // acc init = xz tile (bias already folded in)
  v8f acc;
#pragma unroll
  for (int r = 0; r < 8; ++r) {
    const int bb = m0 + r + 8 * hi;
    acc[r] = xz[((size_t)bb * Tq + t) * Gq + col0 + n];
  }
  __syncthreads();

  // z += h_prev @ R  (K = 256)
  for (int k = 0; k < Uq; k += 4) {
    v2f a = load_a_lds(hA, Uq, k, lane);
    v2f b = load_b_frag(R, Gq, k, col0, lane);
    acc = wmma_f32(a, b, acc);
  }

#pragma unroll
  for (int r = 0; r < 8; ++r)
    zsh[gate][r + 8 * hi][n] = acc[r];
  __syncthreads();

  // Fused gate math: 256 elements over 128 threads.
  for (int e = threadIdx.x; e < 256; e += 128) {
    const int M = e >> 4, N = e & 15;
    const float zi = zsh[0][M][N];
    const float zf = zsh[1][M][N];
    const float zg = zsh[2][M][N];
    const float zo = zsh[3][M][N];
    const float ig = 1.0f / (1.0f + __expf(-zi));
    const float fg = 1.0f / (1.0f + __expf(-zf));
    const float gg = tanhf(zg);
    const float og = 1.0f / (1.0f + __expf(-zo));
    const size_t idx = (size_t)(m0 + M) * Uq + (u0 + N);
    const float cn = fg * c[idx] + ig * gg;
    c[idx] = cn;
    const float hn = og * tanhf(cn);
    h_next[idx] = hn;
    h_seq[((size_t)(m0 + M) * Tq + t) * Uq + (u0 + N)] = hn;
  }
}

// ---- Kernel 3: dense head, wave-per-output --------------------------------
__global__ __launch_bounds__(128) void dense_kernel(
    const float* __restrict__ h,    // [B*T, U]
    const float* __restrict__ Wd,   // [U]
    const float* __restrict__ bd,   // [1]
    float* __restrict__ out) {      // [B*T]
  const int wave = threadIdx.x >> 5;
  const int lane = threadIdx.x & 31;
  const int nrow = blockIdx.x * 4 + wave;
  const float* hp = h + (size_t)nrow * Uq;
  float s = 0.0f;
  for (int j = lane; j < Uq; j += 32) s += hp[j] * Wd[j];
#pragma unroll
  for (int off = 16; off; off >>= 1) s += __shfl_down(s, off, 32);
  if (lane == 0) out[nrow] = s + bd[0];
}

// ---- zero-fill (graph-capture-safe init of h/c state) ---------------------
__global__ void zero_kernel(float* __restrict__ p, int nelem) {
  const int i = blockIdx.x * 256 + threadIdx.x;
  if (i < nelem) p[i] = 0.0f;
}

// ---------------------------------------------------------------------------
extern "C" void kernel_launch(void* const* d_in, const int* in_sizes, int n_in,
                              void* d_out, int out_size, void* d_ws, size_t ws_size,
                              hipStream_t stream) {
  (void)in_sizes; (void)n_in; (void)out_size; (void)ws_size;

  const float* x  = (const float*)d_in[0];
  const float* W[3]  = {(const float*)d_in[1], (const float*)d_in[4], (const float*)d_in[7]};
  const float* Rm[3] = {(const float*)d_in[2], (const float*)d_in[5], (const float*)d_in[8]};
  const float* bs[3] = {(const float*)d_in[3], (const float*)d_in[6], (const float*)d_in[9]};
  const float* Wd = (const float*)d_in[10];
  const float* bd = (const float*)d_in[11];
  float* out = (float*)d_out;

  float* ws    = (float*)d_ws;
  float* xz    = ws;                                  // B*T*G
  float* hseqA = xz    + (size_t)Bq * Tq * Gq;        // B*T*U
  float* hseqB = hseqA + (size_t)Bq * Tq * Uq;        // B*T*U
  float* hping = hseqB + (size_t)Bq * Tq * Uq;        // B*U
  float* hpong = hping + (size_t)Bq * Uq;             // B*U
  float* cbuf  = hpong + (size_t)Bq * Uq;             // B*U  (contiguous w/ hping)

  const float* inseq = x;
  int Din = 128;
  float* outseq[3] = {hseqA, hseqB, hseqA};

  for (int l = 0; l < 3; ++l) {
    // xz = inseq @ W[l] + b[l]
    gemm_xz_kernel<<<dim3((Bq * Tq) / 16, Gq / 64), 128, 0, stream>>>(
        inseq, W[l], bs[l], xz, Din);

    // h0 = c0 = 0  (zeros hping, hpong, cbuf — contiguous)
    zero_kernel<<<(3 * Bq * Uq + 255) / 256, 256, 0, stream>>>(hping, 3 * Bq * Uq);

    for (int t = 0; t < Tq; ++t) {
      float* hp = (t & 1) ? hpong : hping;
      float* hn = (t & 1) ? hping : hpong;
      lstm_step_kernel<<<dim3(Bq / 16, Uq / 16), 128, 0, stream>>>(
          xz, Rm[l], hp, hn, cbuf, outseq[l], t);
    }
    inseq = outseq[l];
    Din = Uq;
  }

  dense_kernel<<<(Bq * Tq) / 4, 128, 0, stream>>>(outseq[2], Wd, bd, out);
}